// context_based_attention_63745904607319
// MI455X (gfx1250) — compile-verified
//
#include <hip/hip_runtime.h>
#include <hip/hip_bf16.h>
#include <math.h>

typedef __attribute__((ext_vector_type(2))) float v2f;
typedef __attribute__((ext_vector_type(8))) float v8f;

#define CCH 128            // channels
#define NODES_PER_WAVE 128 // contiguous nodes each wave owns in the segmented passes

// ---------------------------------------------------------------- zero fill
__global__ void zero_f32(float* __restrict__ p, long n) {
  long i = (long)blockIdx.x * blockDim.x + threadIdx.x;
  long stride = (long)gridDim.x * blockDim.x;
  for (; i < n; i += stride) p[i] = 0.0f;
}

// ------------------------------------------- pass 1: segment sums + counts
// batch is sorted: each wave walks a contiguous node range, accumulates its
// 4 channels (lane*4 .. lane*4+3) in registers, flushes with atomics only
// when the segment id changes (~N/B = 40 nodes/segment -> ~4 flushes/wave).
__global__ void seg_sum_count(const float* __restrict__ x,
                              const int* __restrict__ batch,
                              float* __restrict__ sums,
                              float* __restrict__ counts,
                              int n) {
  const int wave = (blockIdx.x * blockDim.x + threadIdx.x) >> 5;
  const int lane = threadIdx.x & 31;
  long start = (long)wave * NODES_PER_WAVE;
  if (start >= n) return;
  long end = start + NODES_PER_WAVE;
  if (end > n) end = n;

  float4 acc = make_float4(0.f, 0.f, 0.f, 0.f);
  float cnt = 0.f;
  int cur = batch[start];

  for (long i = start; i < end; ++i) {
    int b = batch[i];                       // uniform across the wave
    if (b != cur) {
      float* s = sums + (long)cur * CCH + lane * 4;
      atomicAdd(s + 0, acc.x);
      atomicAdd(s + 1, acc.y);
      atomicAdd(s + 2, acc.z);
      atomicAdd(s + 3, acc.w);
      if (lane == 0) atomicAdd(counts + cur, cnt);
      acc = make_float4(0.f, 0.f, 0.f, 0.f);
      cnt = 0.f;
      cur = b;
    }
    const float4 v = *(const float4*)(x + i * CCH + lane * 4);
    acc.x += v.x; acc.y += v.y; acc.z += v.z; acc.w += v.w;
    cnt += 1.f;
  }
  float* s = sums + (long)cur * CCH + lane * 4;
  atomicAdd(s + 0, acc.x);
  atomicAdd(s + 1, acc.y);
  atomicAdd(s + 2, acc.z);
  atomicAdd(s + 3, acc.w);
  if (lane == 0) atomicAdd(counts + cur, cnt);
}

// --------------------------- pass 2: c = tanh( (sums/count) @ W ) via WMMA
// One wave (32 threads) per 16x16 output tile; K-loop of 32
// v_wmma_f32_16x16x4_f32 instructions. fp32 end to end (matches reference).
// A lane layout (16x4 f32):  M = lane%16, K = elem + 2*(lane/16)
// B lane layout (4x16 f32):  N = lane%16, K = elem + 2*(lane/16)
// D lane layout (16x16 f32): N = lane%16, M = elem + 8*(lane/16)
__global__ void ctx_gemm_tanh(const float* __restrict__ sums,
                              const float* __restrict__ counts,
                              const float* __restrict__ w,
                              float* __restrict__ c,
                              int bseg) {
  const int lane = threadIdx.x;       // 32 threads, full wave, no divergence
  const int l16  = lane & 15;
  const int half = lane >> 4;         // 0 or 1
  const int m0 = blockIdx.x * 16;
  const int n0 = blockIdx.y * 16;

  int mrow = m0 + l16;
  if (mrow > bseg - 1) mrow = bseg - 1;       // clamp (no lane divergence)
  const float inv = 1.0f / fmaxf(counts[mrow], 1.0f);
  const float* arow = sums + (long)mrow * CCH;

  v8f acc = {};
#pragma unroll
  for (int k = 0; k < CCH; k += 4) {
    v2f a, b;
    a.x = arow[k + 2 * half + 0] * inv;
    a.y = arow[k + 2 * half + 1] * inv;
    b.x = w[(k + 2 * half + 0) * CCH + n0 + l16];
    b.y = w[(k + 2 * half + 1) * CCH + n0 + l16];
    acc = __builtin_amdgcn_wmma_f32_16x16x4_f32(
        /*neg_a=*/false, a, /*neg_b=*/false, b,
        /*c_mod=*/(short)0, acc, /*reuse_a=*/false, /*reuse_b=*/false);
  }

#pragma unroll
  for (int j = 0; j < 8; ++j) {
    int m = m0 + j + 8 * half;
    if (m < bseg) c[(long)m * CCH + n0 + l16] = tanhf(acc[j]);
  }
}

// ---------------- pass 3: gate = sigmoid(<x_i, c[batch_i]>); h += gate*x_i
// Same segmented-accumulation trick; c rows (25.6 MB) live in the 192 MB L2.
__global__ void gate_pool(const float* __restrict__ x,
                          const int* __restrict__ batch,
                          const float* __restrict__ c,
                          float* __restrict__ h,
                          int n) {
  const int wave = (blockIdx.x * blockDim.x + threadIdx.x) >> 5;
  const int lane = threadIdx.x & 31;
  long start = (long)wave * NODES_PER_WAVE;
  if (start >= n) return;
  long end = start + NODES_PER_WAVE;
  if (end > n) end = n;

  int cur = batch[start];
  float4 crow = *(const float4*)(c + (long)cur * CCH + lane * 4);
  float4 acc = make_float4(0.f, 0.f, 0.f, 0.f);

  for (long i = start; i < end; ++i) {
    int b = batch[i];                       // uniform across the wave
    if (b != cur) {
      float* hp = h + (long)cur * CCH + lane * 4;
      atomicAdd(hp + 0, acc.x);
      atomicAdd(hp + 1, acc.y);
      atomicAdd(hp + 2, acc.z);
      atomicAdd(hp + 3, acc.w);
      acc = make_float4(0.f, 0.f, 0.f, 0.f);
      cur = b;
      crow = *(const float4*)(c + (long)cur * CCH + lane * 4);
    }
    const float4 v = *(const float4*)(x + i * CCH + lane * 4);
    float pd = v.x * crow.x + v.y * crow.y + v.z * crow.z + v.w * crow.w;
    // wave32 all-reduce of the partial dot (5 xor shuffles)
#pragma unroll
    for (int off = 16; off > 0; off >>= 1) pd += __shfl_xor(pd, off, 32);
    const float gate = 1.0f / (1.0f + expf(-pd));
    acc.x += gate * v.x;
    acc.y += gate * v.y;
    acc.z += gate * v.z;
    acc.w += gate * v.w;
  }
  float* hp = h + (long)cur * CCH + lane * 4;
  atomicAdd(hp + 0, acc.x);
  atomicAdd(hp + 1, acc.y);
  atomicAdd(hp + 2, acc.z);
  atomicAdd(hp + 3, acc.w);
}

// --------------------------------------------------------------- launcher
extern "C" void kernel_launch(void* const* d_in, const int* in_sizes, int n_in,
                              void* d_out, int out_size, void* d_ws, size_t ws_size,
                              hipStream_t stream) {
  const float* x       = (const float*)d_in[0];  // [N,128]
  const float* weightc = (const float*)d_in[1];  // [128,128]
  const int*   batch   = (const int*)d_in[2];    // [N] sorted
  float*       h       = (float*)d_out;          // [B,128]

  const int n    = in_sizes[0] / CCH;            // N nodes
  const int bseg = out_size / CCH;               // B segments

  // workspace layout: sums [B,128] | counts [B] | c [B,128]
  float* sums   = (float*)d_ws;
  float* counts = sums + (long)bseg * CCH;
  float* cbuf   = counts + bseg;

  // zero accumulators (d_out is poisoned by the harness)
  {
    long nz = (long)bseg * CCH;
    zero_f32<<<2048, 256, 0, stream>>>(sums, nz);
    zero_f32<<<256,  256, 0, stream>>>(counts, (long)bseg);
    zero_f32<<<2048, 256, 0, stream>>>(h, nz);
  }

  // pass 1: segment sums + counts
  {
    long waves  = ((long)n + NODES_PER_WAVE - 1) / NODES_PER_WAVE;
    long blocks = (waves * 32 + 255) / 256;
    seg_sum_count<<<(int)blocks, 256, 0, stream>>>(x, batch, sums, counts, n);
  }

  // pass 2: c = tanh(mean @ W), one wave per 16x16 tile
  {
    dim3 grid((bseg + 15) / 16, CCH / 16);
    ctx_gemm_tanh<<<grid, 32, 0, stream>>>(sums, counts, weightc, cbuf, bseg);
  }

  // pass 3: sigmoid gate + gated segment sum
  {
    long waves  = ((long)n + NODES_PER_WAVE - 1) / NODES_PER_WAVE;
    long blocks = (waves * 32 + 255) / 256;
    gate_pool<<<(int)blocks, 256, 0, stream>>>(x, batch, cbuf, h, n);
  }
}